// DenseCapsule_25941602468317
// MI455X (gfx1250) — compile-verified
//
#include <hip/hip_runtime.h>
#include <math.h>

typedef float v2f __attribute__((ext_vector_type(2)));
typedef float v8f __attribute__((ext_vector_type(8)));

#define B_TOT 256
#define I_TOT 1152
#define DIN   8
#define O_TOT 10
#define DOUT  16
#define NC    16                    // i-chunks
#define I_CHUNK (I_TOT / NC)        // 72
#define BT    16                    // batch tile (WMMA N)
#define NBT   (B_TOT / BT)          // 16
#define THREADS 256
#define I_PER_WAVE (I_CHUNK / 8)    // 9

// Routing pass. Recomputes x_hat[d][b] tiles via V_WMMA_F32_16X16X4_F32 and:
//  UNIFORM=1: s_part = 0.1 * sum_i x_hat           (iteration 0, c uniform)
//  UNIFORM=0: logits = vsum . x_hat; c = softmax_o; s_part = sum_i c*x_hat
template <int UNIFORM>
__global__ __launch_bounds__(THREADS) void caps_pass(
    const float* __restrict__ x, const float* __restrict__ w,
    const float* __restrict__ vsum, float* __restrict__ s_part) {
  // aliased LDS: phase 1 = x tile transposed (9216 f), phase 2 = 4 reduction slots (10240 f)
  __shared__ float smem[10240];

  const int bt = blockIdx.x;            // batch tile 0..15
  const int ic = blockIdx.y;            // i chunk   0..15
  const int b0 = bt * BT;
  const int i0 = ic * I_CHUNK;
  const int tid  = threadIdx.x;
  const int lane = tid & 31;
  const int wv   = tid >> 5;
  const int bl   = lane & 15;           // WMMA: batch col / d row index
  const int h    = lane >> 4;           // lane-half selects K-pair / d-half

  // Stage x tile transposed: x[b0+b][i0+il][c] -> smem[il*128 + c*16 + b]
  for (int idx = tid; idx < BT * I_CHUNK * DIN; idx += THREADS) {
    int b  = idx / (I_CHUNK * DIN);
    int r  = idx % (I_CHUNK * DIN);
    int il = r / DIN;
    int c  = r % DIN;
    smem[il * (DIN * BT) + c * BT + b] =
        x[(size_t)(b0 + b) * (I_TOT * DIN) + (size_t)(i0 + il) * DIN + c];
  }
  __syncthreads();

  // Preload vsum fragment in x_hat D-layout: element k -> (b=bl, d=k+8h)
  v8f vs[O_TOT];
  if (!UNIFORM) {
#pragma unroll
    for (int o = 0; o < O_TOT; ++o)
#pragma unroll
      for (int k = 0; k < 8; ++k)
        vs[o][k] = vsum[(size_t)(b0 + bl) * (O_TOT * DOUT) + o * DOUT + (k + 8 * h)];
  }

  v8f sacc[O_TOT];
#pragma unroll
  for (int o = 0; o < O_TOT; ++o) sacc[o] = (v8f){};

  for (int t = 0; t < I_PER_WAVE; ++t) {
    const int il = wv * I_PER_WAVE + t;
    const int ig = i0 + il;

    // B fragments (x slice, K=c x N=b), K = r + 2h per 4-wide frag
    const float* xcol = &smem[il * (DIN * BT)];
    v2f xb0, xb1;
    xb0[0] = xcol[(2 * h + 0) * BT + bl];
    xb0[1] = xcol[(2 * h + 1) * BT + bl];
    xb1[0] = xcol[(4 + 2 * h + 0) * BT + bl];
    xb1[1] = xcol[(4 + 2 * h + 1) * BT + bl];

    v8f   xh[O_TOT];
    float lg[O_TOT];
#pragma unroll
    for (int o = 0; o < O_TOT; ++o) {
      // A fragment: W[o,ig,d=bl,c], c = kk*4 + r + 2h  (contiguous float2)
      const float* wp = &w[((size_t)(o * I_TOT + ig) * DOUT + bl) * DIN];
      v2f wa0 = *(const v2f*)(wp + 2 * h);
      v2f wa1 = *(const v2f*)(wp + 4 + 2 * h);
      v8f acc = (v8f){};
      acc = __builtin_amdgcn_wmma_f32_16x16x4_f32(false, wa0, false, xb0,
                                                  (short)0, acc, false, false);
      acc = __builtin_amdgcn_wmma_f32_16x16x4_f32(false, wa1, false, xb1,
                                                  (short)0, acc, false, false);
      xh[o] = acc;
      if (!UNIFORM) {
        float lp = 0.f;
#pragma unroll
        for (int k = 0; k < 8; ++k) lp += vs[o][k] * acc[k];
        lg[o] = lp;
      }
    }

    if (UNIFORM) {
#pragma unroll
      for (int o = 0; o < O_TOT; ++o) sacc[o] += xh[o];
    } else {
      // complete the d-dot across lane halves; both halves now hold full logits
#pragma unroll
      for (int o = 0; o < O_TOT; ++o) lg[o] += __shfl_xor(lg[o], 16, 32);
      float m = lg[0];
#pragma unroll
      for (int o = 1; o < O_TOT; ++o) m = fmaxf(m, lg[o]);
      float e[O_TOT];
      float den = 0.f;
#pragma unroll
      for (int o = 0; o < O_TOT; ++o) { e[o] = __expf(lg[o] - m); den += e[o]; }
      float inv = 1.f / den;
#pragma unroll
      for (int o = 0; o < O_TOT; ++o) {
        float co = e[o] * inv;
#pragma unroll
        for (int k = 0; k < 8; ++k) sacc[o][k] += co * xh[o][k];
      }
    }
  }

  // Deterministic staged cross-wave reduction (4 LDS slots, fixed order)
  __syncthreads();
  if (wv < 4) {
#pragma unroll
    for (int o = 0; o < O_TOT; ++o)
#pragma unroll
      for (int k = 0; k < 8; ++k)
        smem[wv * 2560 + o * 256 + k * 32 + lane] = sacc[o][k];
  }
  __syncthreads();
  if (wv >= 4) {
#pragma unroll
    for (int o = 0; o < O_TOT; ++o)
#pragma unroll
      for (int k = 0; k < 8; ++k)
        smem[(wv - 4) * 2560 + o * 256 + k * 32 + lane] += sacc[o][k];
  }
  __syncthreads();
  const float scale = UNIFORM ? 0.1f : 1.0f;
  for (int j = tid; j < 2560; j += THREADS) {
    float s = smem[j] + smem[2560 + j] + smem[5120 + j] + smem[7680 + j];
    int o = j >> 8;
    int k = (j >> 5) & 7;
    int l = j & 31;
    int b = l & 15;
    int d = k + 8 * (l >> 4);
    s_part[(((size_t)ic * B_TOT + (b0 + b)) * O_TOT + o) * DOUT + d] = s * scale;
  }
}

// Sum partials over i-chunks, squash, update vsum / emit output.
// mode 0: vsum = v ; mode 1: vsum += v ; mode 2: out = v
__global__ __launch_bounds__(256) void caps_squash(
    const float* __restrict__ s_part, float* __restrict__ vsum,
    float* __restrict__ out, int mode) {
  int gid = blockIdx.x * 256 + threadIdx.x;  // 0..40959
  int d  = gid & 15;
  int bo = gid >> 4;                         // b*10 + o
  float s = 0.f;
  for (int icc = 0; icc < NC; ++icc)
    s += s_part[((size_t)icc * B_TOT * O_TOT + bo) * DOUT + d];
  float msq = s * s;                         // reduce over 16-lane d-group
  msq += __shfl_xor(msq, 1, 32);
  msq += __shfl_xor(msq, 2, 32);
  msq += __shfl_xor(msq, 4, 32);
  msq += __shfl_xor(msq, 8, 32);
  float mag = sqrtf(msq);
  float v = s * (msq / ((1.f + msq) * (mag + 1e-8f)));
  if (mode == 2)      out[(size_t)bo * DOUT + d] = v;
  else if (mode == 1) vsum[(size_t)bo * DOUT + d] += v;
  else                vsum[(size_t)bo * DOUT + d] = v;
}

extern "C" void kernel_launch(void* const* d_in, const int* in_sizes, int n_in,
                              void* d_out, int out_size, void* d_ws, size_t ws_size,
                              hipStream_t stream) {
  (void)in_sizes; (void)n_in; (void)out_size; (void)ws_size;
  const float* x = (const float*)d_in[0];      // [256,1152,8]
  const float* w = (const float*)d_in[1];      // [10,1152,16,8]
  float* out    = (float*)d_out;               // [256,10,16]
  float* s_part = (float*)d_ws;                                   // NC*B*O*D floats
  float* vsum   = s_part + (size_t)NC * B_TOT * O_TOT * DOUT;     // B*O*D floats

  dim3 grid(NBT, NC);
  // iter 0: uniform c
  caps_pass<1><<<grid, THREADS, 0, stream>>>(x, w, vsum, s_part);
  caps_squash<<<160, 256, 0, stream>>>(s_part, vsum, out, 0);
  // iter 1
  caps_pass<0><<<grid, THREADS, 0, stream>>>(x, w, vsum, s_part);
  caps_squash<<<160, 256, 0, stream>>>(s_part, vsum, out, 1);
  // iter 2 (final)
  caps_pass<0><<<grid, THREADS, 0, stream>>>(x, w, vsum, s_part);
  caps_squash<<<160, 256, 0, stream>>>(s_part, vsum, out, 2);
}